// GCN_Module_14010183320164
// MI455X (gfx1250) — compile-verified
//
#include <hip/hip_runtime.h>
#include <hip/hip_bf16.h>

// CDNA5 / gfx1250, wave32. V_WMMA_F32_16X16X4_F32 for all big GEMMs,
// 5x2 register-blocked (10 accumulators / wave) with double-buffered loads.

typedef float v2f __attribute__((ext_vector_type(2)));
typedef float v8f __attribute__((ext_vector_type(8)));

constexpr int kNFR  = 256;     // projection dim
constexpr int kNFG  = 1000;    // feature dim
constexpr int kNG   = 4;       // groups
constexpr int kN    = 5;       // boxes per batch item
constexpr int kBT   = 8192;    // batch*time
constexpr int kBTile= 16;      // b's per block
constexpr int kRows = kBTile * kN;   // 80 rows = 5 exact 16-row WMMA tiles
constexpr int kEPad = 1008;    // 63 col tiles of 16 (1000 padded)
constexpr int kTHP  = 264;     // LDS pitch for theta/phi (bank spread)
constexpr float kLnEps = 1e-5f;

static __device__ __forceinline__ v8f wmma_f32_k4(v2f a, v2f b, v8f c) {
  // D = A(16x4 f32) x B(4x16 f32) + C(16x16 f32)
  return __builtin_amdgcn_wmma_f32_16x16x4_f32(false, a, false, b, (short)0, c,
                                               false, false);
}

// ---------------------------------------------------------------------------
// Kernel 1: theta/phi projections (WMMA), 5x5 similarity, softmax -> rel
// grid = (NG, BT/kBTile), block = 256 (8 waves)
// ---------------------------------------------------------------------------
__global__ __launch_bounds__(256) void rel_kernel(
    const float* __restrict__ x,        // [BT, 5, NFG]
    const float* __restrict__ theta_w,  // [G, NFR, NFG]
    const float* __restrict__ theta_b,  // [G, NFR]
    const float* __restrict__ phi_w,    // [G, NFR, NFG]
    const float* __restrict__ phi_b,    // [G, NFR]
    float* __restrict__ rel,            // ws: [G, BT*5, 5]
    float* __restrict__ rel_out)        // d_out tail: [BT*5, 5] (group 3)
{
  extern __shared__ float smem[];
  float* th_s  = smem;                         // [80][264]
  float* ph_s  = th_s + kRows * kTHP;          // [80][264]
  float* sim_s = ph_s + kRows * kTHP;          // [80][8]

  const int g    = blockIdx.x;
  const int bt   = blockIdx.y;
  const int tid  = threadIdx.x;
  const int lane = tid & 31;
  const int wave = tid >> 5;
  const long rowbase = (long)bt * kRows;       // global row = b*5+n
  const float* xblk  = x + rowbase * kNFG;

  const int mrow = lane & 15;                  // row/col within a 16-tile
  const int koff = (lane >> 4) * 2;            // K sub-offset for this lane

  // --- WMMA phase: 16 super-tiles (8 theta col-pairs + 8 phi col-pairs),
  //     each = 5 rowtiles x 2 coltiles, K = 1000 ---
  for (int u = wave; u < 16; u += 8) {
    const bool isphi = (u >= 8);
    const int  ct0 = (isphi ? u - 8 : u) * 2;  // first of two coltiles (f)
    const float* W = (isphi ? phi_w : theta_w) + (long)g * kNFR * kNFG;

    const float* ap[5];
#pragma unroll
    for (int r = 0; r < 5; ++r)
      ap[r] = xblk + (long)(r * 16 + mrow) * kNFG + koff;
    const float* bp[2];
#pragma unroll
    for (int c = 0; c < 2; ++c)
      bp[c] = W + (long)((ct0 + c) * 16 + mrow) * kNFG + koff;

    v8f acc[10];
#pragma unroll
    for (int i = 0; i < 10; ++i) acc[i] = (v8f){0.f,0.f,0.f,0.f,0.f,0.f,0.f,0.f};

    v2f a_cur[5], b_cur[2], a_nxt[5], b_nxt[2];
#pragma unroll
    for (int r = 0; r < 5; ++r) { a_cur[r].x = ap[r][0]; a_cur[r].y = ap[r][1]; }
#pragma unroll
    for (int c = 0; c < 2; ++c) { b_cur[c].x = bp[c][0]; b_cur[c].y = bp[c][1]; }

#pragma unroll 2
    for (int k = 0; k < kNFG; k += 4) {
      const int kn = k + 4;
      if (kn < kNFG) {
#pragma unroll
        for (int r = 0; r < 5; ++r) { a_nxt[r].x = ap[r][kn]; a_nxt[r].y = ap[r][kn + 1]; }
#pragma unroll
        for (int c = 0; c < 2; ++c) { b_nxt[c].x = bp[c][kn]; b_nxt[c].y = bp[c][kn + 1]; }
      }
#pragma unroll
      for (int r = 0; r < 5; ++r)
#pragma unroll
        for (int c = 0; c < 2; ++c)
          acc[r * 2 + c] = wmma_f32_k4(a_cur[r], b_cur[c], acc[r * 2 + c]);
#pragma unroll
      for (int r = 0; r < 5; ++r) a_cur[r] = a_nxt[r];
#pragma unroll
      for (int c = 0; c < 2; ++c) b_cur[c] = b_nxt[c];
    }

    float* dst = isphi ? ph_s : th_s;
    const int row0 = (lane >> 4) << 3;         // 0 or 8 within tile
#pragma unroll
    for (int c = 0; c < 2; ++c) {
      const int col = (ct0 + c) * 16 + mrow;
#pragma unroll
      for (int r = 0; r < 5; ++r)
#pragma unroll
        for (int j = 0; j < 8; ++j)
          dst[(r * 16 + row0 + j) * kTHP + col] = acc[r * 2 + c][j];
    }
  }
  __syncthreads();

  // --- add biases ---
  for (int idx = tid; idx < kRows * kNFR; idx += 256) {
    const int r = idx >> 8, f = idx & 255;
    th_s[r * kTHP + f] += theta_b[g * kNFR + f];
    ph_s[r * kTHP + f] += phi_b[g * kNFR + f];
  }
  __syncthreads();

  // --- sim[b][n][m] = theta[b,n,:] . phi[b,m,:] / 16 ---
  for (int idx = tid; idx < kBTile * kN * kN; idx += 256) {
    const int b = idx / 25;
    const int n = (idx % 25) / 5;
    const int m = idx % 5;
    const float* tr = th_s + (b * 5 + n) * kTHP;
    const float* pr = ph_s + (b * 5 + m) * kTHP;
    float s = 0.f;
    for (int f = 0; f < kNFR; ++f) s += tr[f] * pr[f];
    sim_s[(b * 5 + n) * 8 + m] = s * (1.0f / 16.0f);  // 1/sqrt(256)
  }
  __syncthreads();

  // --- softmax per row (80 rows x 5) and write out ---
  if (tid < kRows) {
    float* sr = sim_s + tid * 8;
    float mx = sr[0];
#pragma unroll
    for (int m = 1; m < 5; ++m) mx = fmaxf(mx, sr[m]);
    float e[5], sum = 0.f;
#pragma unroll
    for (int m = 0; m < 5; ++m) { e[m] = __expf(sr[m] - mx); sum += e[m]; }
    const float inv = 1.0f / sum;
    const long base = (long)g * kBT * 25 + (rowbase + tid) * 5;
#pragma unroll
    for (int m = 0; m < 5; ++m) {
      const float v = e[m] * inv;
      rel[base + m] = v;
      if (g == kNG - 1) rel_out[(rowbase + tid) * 5 + m] = v;
    }
  }
}

// ---------------------------------------------------------------------------
// Kernel 2: per block (16 b's), loop g: Z = X @ gcn_w[g]^T via WMMA into LDS,
// then h = rel @ Z on the fly -> LayerNorm -> ReLU -> accumulate into out.
// grid = BT/kBTile, block = 256 (8 waves)
// ---------------------------------------------------------------------------
__global__ __launch_bounds__(256) void gcn_kernel(
    const float* __restrict__ x,      // [BT, 5, NFG]
    const float* __restrict__ gcn_w,  // [G, NFG, NFG]
    const float* __restrict__ ln_g,   // [G, NFG]
    const float* __restrict__ ln_b,   // [G, NFG]
    const float* __restrict__ rel,    // ws: [G, BT*5, 5]
    float* __restrict__ out)          // [BT, 5, NFG]
{
  extern __shared__ float smem[];
  float* Zf      = smem;                     // [80][1008]
  float* rel_s   = Zf + kRows * kEPad;       // [80][5]
  float* sum_s   = rel_s + kRows * 5;        // [80]
  float* sumsq_s = sum_s + kRows;            // [80]

  const int bt   = blockIdx.x;
  const int tid  = threadIdx.x;
  const int lane = tid & 31;
  const int wave = tid >> 5;
  const long rowbase = (long)bt * kRows;
  const float* xblk  = x + rowbase * kNFG;

  const int mrow = lane & 15;
  const int koff = (lane >> 4) * 2;

  for (int g = 0; g < kNG; ++g) {
    // load rel coefficients for this tile
    for (int idx = tid; idx < kRows * 5; idx += 256)
      rel_s[idx] = rel[(long)g * kBT * 25 + rowbase * 5 + idx];
    __syncthreads();

    // --- WMMA: Z[80 x 1008] = Xtile @ Gw^T ---
    // 64 padded coltiles as 32 pairs; each wave: 4 pairs x (5x2) super-tile.
    const float* Gw = gcn_w + (long)g * kNFG * kNFG;
    for (int pair = wave; pair < 32; pair += 8) {
      const int ct0 = pair * 2;

      const float* ap[5];
#pragma unroll
      for (int r = 0; r < 5; ++r)
        ap[r] = xblk + (long)(r * 16 + mrow) * kNFG + koff;

      const float* bp[2];
      bool bval[2];
#pragma unroll
      for (int c = 0; c < 2; ++c) {
        const int e = (ct0 + c) * 16 + mrow;
        bval[c] = (e < kNFG);
        bp[c] = Gw + (long)e * kNFG + koff;
      }

      v8f acc[10];
#pragma unroll
      for (int i = 0; i < 10; ++i) acc[i] = (v8f){0.f,0.f,0.f,0.f,0.f,0.f,0.f,0.f};

      v2f a_cur[5], b_cur[2], a_nxt[5], b_nxt[2];
#pragma unroll
      for (int r = 0; r < 5; ++r) { a_cur[r].x = ap[r][0]; a_cur[r].y = ap[r][1]; }
#pragma unroll
      for (int c = 0; c < 2; ++c) {
        if (bval[c]) { b_cur[c].x = bp[c][0]; b_cur[c].y = bp[c][1]; }
        else         { b_cur[c].x = 0.f;      b_cur[c].y = 0.f; }
      }

#pragma unroll 2
      for (int k = 0; k < kNFG; k += 4) {
        const int kn = k + 4;
        if (kn < kNFG) {
#pragma unroll
          for (int r = 0; r < 5; ++r) { a_nxt[r].x = ap[r][kn]; a_nxt[r].y = ap[r][kn + 1]; }
#pragma unroll
          for (int c = 0; c < 2; ++c) {
            if (bval[c]) { b_nxt[c].x = bp[c][kn]; b_nxt[c].y = bp[c][kn + 1]; }
            else         { b_nxt[c].x = 0.f;       b_nxt[c].y = 0.f; }
          }
        }
#pragma unroll
        for (int r = 0; r < 5; ++r)
#pragma unroll
          for (int c = 0; c < 2; ++c)
            acc[r * 2 + c] = wmma_f32_k4(a_cur[r], b_cur[c], acc[r * 2 + c]);
#pragma unroll
        for (int r = 0; r < 5; ++r) a_cur[r] = a_nxt[r];
#pragma unroll
        for (int c = 0; c < 2; ++c) b_cur[c] = b_nxt[c];
      }

      const int row0 = (lane >> 4) << 3;
#pragma unroll
      for (int c = 0; c < 2; ++c) {
        const int ct = ct0 + c;
        if (ct < 63) {                          // uniform per wave; tile 63 is pad
          const int col = ct * 16 + mrow;
#pragma unroll
          for (int r = 0; r < 5; ++r)
#pragma unroll
            for (int j = 0; j < 8; ++j)
              Zf[(r * 16 + row0 + j) * kEPad + col] = acc[r * 2 + c][j];
        }
      }
    }
    __syncthreads();

    // --- per-row sum / sumsq of h = rel @ Z (wave-parallel over e) ---
    for (int rr = 0; rr < 10; ++rr) {
      const int r  = wave * 10 + rr;            // 8 waves * 10 = 80 rows
      const int b5 = (r / 5) * 5;
      const float* rw = rel_s + r * 5;
      float s = 0.f, q = 0.f;
      for (int e = lane; e < kNFG; e += 32) {
        const float h = rw[0] * Zf[(b5 + 0) * kEPad + e]
                      + rw[1] * Zf[(b5 + 1) * kEPad + e]
                      + rw[2] * Zf[(b5 + 2) * kEPad + e]
                      + rw[3] * Zf[(b5 + 3) * kEPad + e]
                      + rw[4] * Zf[(b5 + 4) * kEPad + e];
        s += h; q += h * h;
      }
#pragma unroll
      for (int off = 16; off > 0; off >>= 1) {
        s += __shfl_xor(s, off, 32);
        q += __shfl_xor(q, off, 32);
      }
      if (lane == 0) { sum_s[r] = s; sumsq_s[r] = q; }
    }
    __syncthreads();

    // --- LayerNorm + ReLU + accumulate into out ---
    const float* lg = ln_g + g * kNFG;
    const float* lb = ln_b + g * kNFG;
    for (int idx = tid; idx < kRows * kNFG; idx += 256) {
      const int r = idx / kNFG;
      const int e = idx % kNFG;
      const int b5 = (r / 5) * 5;
      const float* rw = rel_s + r * 5;
      const float h = rw[0] * Zf[(b5 + 0) * kEPad + e]
                    + rw[1] * Zf[(b5 + 1) * kEPad + e]
                    + rw[2] * Zf[(b5 + 2) * kEPad + e]
                    + rw[3] * Zf[(b5 + 3) * kEPad + e]
                    + rw[4] * Zf[(b5 + 4) * kEPad + e];
      const float mu  = sum_s[r] * (1.0f / kNFG);
      const float var = sumsq_s[r] * (1.0f / kNFG) - mu * mu;
      float v = (h - mu) * rsqrtf(var + kLnEps) * lg[e] + lb[e];
      v = fmaxf(v, 0.f);
      const long o = (rowbase + r) * (long)kNFG + e;
      if (g == 0) out[o] = v;
      else        out[o] += v;                  // block owns these rows; no races
    }
    __syncthreads();  // protect Zf / rel_s before next group's writes
  }
}

// ---------------------------------------------------------------------------
extern "C" void kernel_launch(void* const* d_in, const int* in_sizes, int n_in,
                              void* d_out, int out_size, void* d_ws, size_t ws_size,
                              hipStream_t stream) {
  const float* x       = (const float*)d_in[0];
  const float* theta_w = (const float*)d_in[1];
  const float* theta_b = (const float*)d_in[2];
  const float* phi_w   = (const float*)d_in[3];
  const float* phi_b   = (const float*)d_in[4];
  const float* gcn_w   = (const float*)d_in[5];
  const float* ln_g    = (const float*)d_in[6];
  const float* ln_b    = (const float*)d_in[7];

  float* out     = (float*)d_out;
  float* rel_out = out + (size_t)kBT * kN * kNFG;   // tuple tail: relation[-1]
  float* rel     = (float*)d_ws;                    // 4*8192*25 floats = 3.3 MB

  const dim3 blk(256);

  const dim3 g1(kNG, kBT / kBTile);  // (4, 512)
  const size_t sh1 = (size_t)(kRows * kTHP * 2 + kRows * 8) * sizeof(float); // ~171.5 KB
  rel_kernel<<<g1, blk, sh1, stream>>>(x, theta_w, theta_b, phi_w, phi_b, rel, rel_out);

  const dim3 g2(kBT / kBTile);       // 512
  const size_t sh2 = (size_t)(kRows * kEPad + kRows * 5 + 2 * kRows) * sizeof(float); // ~324.8 KB
  gcn_kernel<<<g2, blk, sh2, stream>>>(x, gcn_w, ln_g, ln_b, rel, out);
}